// NLLayer_36309653520599
// MI455X (gfx1250) — compile-verified
//
#include <hip/hip_runtime.h>

typedef __attribute__((ext_vector_type(2))) float v2f;
typedef __attribute__((ext_vector_type(8))) float v8f;

#define BATCH   32
#define NATOMS  1024
#define TILES_J (NATOMS / 16)   // 64 j-tiles of 16 columns

#if defined(__gfx1250__) && __has_builtin(__builtin_amdgcn_tensor_load_to_lds)
#define USE_TDM 1
typedef __attribute__((ext_vector_type(4))) unsigned int v4u32;
typedef __attribute__((ext_vector_type(8))) int          v8i32;
typedef __attribute__((ext_vector_type(4))) int          v4i32;
#else
#define USE_TDM 0
#endif

// One block per (frame b, 16-row stripe i0). 256 threads = 8 waves (wave32).
//
// Staging: the Tensor Data Mover DMAs the whole frame's positions
// (3072 f32 = 12 KB) into LDS as a 3072x1 tile (no VGPR round-trip),
// tracked with TENSORcnt; fallback is float4 global->VGPR->ds staging.
//
// Compute: each wave produces 16x16 pair tiles with V_WMMA_F32_16X16X4_F32:
//   D[m][n] = A[m][0]*B[0][n] + A[m][1]*B[1][n] = x_i[m]*1 + 1*(-x_j[n])
// then applies the minimum-image wrap d -= rint(d/box)*box (v_rndne + pk_fma)
// and stores one float3 per (i,j): lanes 0..15 cover consecutive j, giving
// contiguous 192B b96 store spans per half-wave (store-bandwidth-bound).
__global__ __launch_bounds__(256) void rij_min_image_kernel(
    const float* __restrict__ pos,    // (B, N, 3)
    const float* __restrict__ cell,   // (B, 3, 3) diagonal cells
    float* __restrict__ out)          // (B, N, N, 3)
{
    __shared__ float spos[NATOMS * 3];   // 12 KB: whole frame staged once

    const int b   = blockIdx.y;
    const int i0  = blockIdx.x << 4;     // 16-row stripe
    const int tid = threadIdx.x;

#if USE_TDM
    // Wave 0 issues one TDM descriptor for the whole 12 KB frame tile.
    // (TDM ignores EXEC; issuing from a single wave avoids 8x duplication.)
    if (tid < 32) {
        const unsigned long long ga =
            (unsigned long long)(uintptr_t)(pos + (size_t)b * NATOMS * 3);
        // Generic pointer to LDS carries the LDS byte offset in its low 32b.
        const unsigned int lds_addr = (unsigned int)(uintptr_t)spos;

        // D# group 0: count=1 (valid user descriptor), lds_addr,
        // global_addr[56:0], type=2 ("image") in bits 127:126.
        v4u32 g0 = { 1u,
                     lds_addr,
                     (unsigned int)ga,
                     ((unsigned int)(ga >> 32) & 0x01FFFFFFu) | (2u << 30) };

        // D# group 1: wg_mask=0, data_size=2 (4 bytes); 2-D tensor
        // 3072x1, tile 3072x1, dim0 stride 3072 (element units).
        v8i32 g1 = { (int)(2u << 16),                  // data_size=4B
                     (int)((3072u & 0xFFFFu) << 16),   // tensor_dim0 lo16 @ bits63:48
                     (int)(1u << 16),                  // tensor_dim0 hi=0 | tensor_dim1=1
                     (int)((3072u & 0xFFFFu) << 16),   // tensor_dim1 hi=0 | tile_dim0=3072
                     1,                                // tile_dim1=1, tile_dim2=0
                     3072,                             // tensor_dim0_stride lo32
                     (int)((3072u & 0xFFFFu) << 16),   // stride0 hi=0 | dim1_stride lo16
                     0 };                              // dim1_stride hi

        v4i32 gz = { 0, 0, 0, 0 };                     // groups 2/3 unused (2-D)
        v8i32 g5 = { 0, 0, 0, 0, 0, 0, 0, 0 };        // 5th operand (unused), zero

        // 6-arg form (this toolchain): (g0, g1, g2, g3, g5, cpol)
        __builtin_amdgcn_tensor_load_to_lds(g0, g1, gz, gz, g5, 0);
        __builtin_amdgcn_s_wait_tensorcnt(0);          // s_wait_tensorcnt 0x0
    }
#else
    // Fallback cooperative stage: 3072 floats = 768 x b128.
    {
        const float4* gp = (const float4*)(pos + (size_t)b * NATOMS * 3);
        float4*       sp = (float4*)spos;
#pragma unroll
        for (int k = 0; k < 3; ++k)
            sp[tid + 256 * k] = gp[tid + 256 * k];
    }
#endif

    // Diagonal (cubic) cell for this frame.
    const float bx = cell[(size_t)b * 9 + 0];
    const float by = cell[(size_t)b * 9 + 4];
    const float bz = cell[(size_t)b * 9 + 8];
    const float ibx = 1.0f / bx, iby = 1.0f / by, ibz = 1.0f / bz;

    __syncthreads();   // publish LDS tile to all 8 waves

    const int wave = tid >> 5;           // 8 waves per block
    const int lane = tid & 31;           // wave32
    const int l16  = lane & 15;
    const bool lo  = (lane < 16);

    // A-matrix (16x4 f32, per component): lanes 0-15 hold K0/K1 = {x_i, 1},
    // lanes 16-31 hold K2/K3 = {0, 0}. Rows M = lane (lo half).
    v2f aX = lo ? (v2f){spos[(i0 + l16) * 3 + 0], 1.0f} : (v2f){0.0f, 0.0f};
    v2f aY = lo ? (v2f){spos[(i0 + l16) * 3 + 1], 1.0f} : (v2f){0.0f, 0.0f};
    v2f aZ = lo ? (v2f){spos[(i0 + l16) * 3 + 2], 1.0f} : (v2f){0.0f, 0.0f};

    // D-matrix layout: VGPR v holds M=v (lanes 0-15) and M=v+8 (lanes 16-31);
    // column N = lane&15.
    const int irow_base = i0 + ((lane >> 4) << 3);

    for (int jt = wave; jt < TILES_J; jt += 8) {
        const int j0 = jt << 4;

        // B-matrix (4x16 f32): row K striped across lanes within a VGPR;
        // lanes 0-15 carry K0/K1 = {1, -x_j}, lanes 16-31 carry K2/K3 = 0.
        v2f bX = lo ? (v2f){1.0f, -spos[(j0 + l16) * 3 + 0]} : (v2f){0.0f, 0.0f};
        v2f bY = lo ? (v2f){1.0f, -spos[(j0 + l16) * 3 + 1]} : (v2f){0.0f, 0.0f};
        v2f bZ = lo ? (v2f){1.0f, -spos[(j0 + l16) * 3 + 2]} : (v2f){0.0f, 0.0f};

        v8f dX, dY, dZ;
#if __has_builtin(__builtin_amdgcn_wmma_f32_16x16x4_f32)
        {
            v8f c0 = {};
            // 8 args: (neg_a, A, neg_b, B, c_mod, C, reuse_a, reuse_b)
            dX = __builtin_amdgcn_wmma_f32_16x16x4_f32(false, aX, false, bX,
                                                       (short)0, c0, false, false);
            dY = __builtin_amdgcn_wmma_f32_16x16x4_f32(false, aY, false, bY,
                                                       (short)0, c0, false, false);
            dZ = __builtin_amdgcn_wmma_f32_16x16x4_f32(false, aZ, false, bZ,
                                                       (short)0, c0, false, false);
        }
#else
        // Scalar fallback with identical (i, j) -> lane/VGPR mapping.
#pragma unroll
        for (int v = 0; v < 8; ++v) {
            const int i = irow_base + v;
            const int j = j0 + l16;
            dX[v] = spos[i * 3 + 0] - spos[j * 3 + 0];
            dY[v] = spos[i * 3 + 1] - spos[j * 3 + 1];
            dZ[v] = spos[i * 3 + 2] - spos[j * 3 + 2];
        }
#endif

        // Minimum-image wrap + coalesced float3 stores.
#pragma unroll
        for (int v = 0; v < 8; ++v) {
            const int i = irow_base + v;
            const int j = j0 + l16;

            float x = dX[v];
            float y = dY[v];
            float z = dZ[v];
            x -= rintf(x * ibx) * bx;   // v_rndne_f32 + fma
            y -= rintf(y * iby) * by;
            z -= rintf(z * ibz) * bz;

            const size_t off = (((size_t)b * NATOMS + i) * NATOMS + j) * 3;
            out[off + 0] = x;
            out[off + 1] = y;
            out[off + 2] = z;
        }
    }
}

extern "C" void kernel_launch(void* const* d_in, const int* in_sizes, int n_in,
                              void* d_out, int out_size, void* d_ws, size_t ws_size,
                              hipStream_t stream) {
    const float* pos  = (const float*)d_in[0];   // (32, 1024, 3) fp32
    const float* cell = (const float*)d_in[1];   // (32, 3, 3)   fp32
    float*       out  = (float*)d_out;           // (32, 1024, 1024, 3) fp32

    (void)in_sizes; (void)n_in; (void)out_size; (void)d_ws; (void)ws_size;

    dim3 grid(NATOMS / 16, BATCH);   // (64, 32): one block per 16-row stripe
    rij_min_image_kernel<<<grid, 256, 0, stream>>>(pos, cell, out);
}